// GE2ELoss_88235808129580
// MI455X (gfx1250) — compile-verified
//
#include <hip/hip_runtime.h>
#include <hip/hip_bf16.h>
#include <math.h>

// GE2E loss, fused WMMA implementation for gfx1250 (MI455X).
// N_SPK=1024, N_UTT=32, D=128 (fixed by the reference).

typedef __attribute__((ext_vector_type(2))) float v2f;
typedef __attribute__((ext_vector_type(8))) float v8f;

#define NSPK 1024
#define NUTT 32
#define DIM  128
#define NROW (NSPK * NUTT)          // 32768 utterance rows
#define EPSF 1e-8f
#define NWAVE (NROW / 16)           // 2048 waves in the main kernel

// ---------------------------------------------------------------------------
// Kernel 1: per-speaker prep.
//   block = 128 threads (4 waves), one speaker per block.
//   Outputs:
//     cent_hat[n][d] = (mean_m e[n][m][d]) / max(||cent||, eps)
//     inv_en[u]      = 1 / max(||e_u||, eps)
//     own_logit[u]   = w * cos(e_u, cent_excl_u) + b   (diagonal logit)
// ---------------------------------------------------------------------------
__global__ __launch_bounds__(128)
void ge2e_prep(const float* __restrict__ emb,
               const float* __restrict__ wp, const float* __restrict__ bp,
               float* __restrict__ cent_hat,
               float* __restrict__ inv_en,
               float* __restrict__ own_logit) {
    __shared__ float etile[NUTT][DIM];   // 16 KB: this speaker's 32x128 block
    __shared__ float cent_s[DIM];        // centroid * NUTT (the sum)
    __shared__ float red4[4];

    const int n    = blockIdx.x;         // speaker
    const int d    = threadIdx.x;        // dimension 0..127
    const int wv   = threadIdx.x >> 5;
    const int lane = threadIdx.x & 31;

    // Phase A: column sums + stage tile in LDS
    float s = 0.0f;
    const float* base = emb + (size_t)n * NUTT * DIM + d;
    #pragma unroll 4
    for (int m = 0; m < NUTT; ++m) {
        float v = base[(size_t)m * DIM];
        etile[m][d] = v;
        s += v;
    }
    cent_s[d] = s;

    // centroid norm via wave shuffle + cross-wave LDS reduce
    float cent = s * (1.0f / (float)NUTT);
    float sq = cent * cent;
    #pragma unroll
    for (int off = 16; off >= 1; off >>= 1) sq += __shfl_xor(sq, off);
    if (lane == 0) red4[wv] = sq;
    __syncthreads();
    float n2   = red4[0] + red4[1] + red4[2] + red4[3];
    float invc = 1.0f / fmaxf(sqrtf(n2), EPSF);
    cent_hat[(size_t)n * DIM + d] = cent * invc;

    // Phase B: each wave handles 8 utterances; lane covers 4 dims.
    const float W = wp[0];
    const float B = bp[0];
    #pragma unroll
    for (int i = 0; i < 8; ++i) {
        int m = wv * 8 + i;
        float dot = 0.0f, e2 = 0.0f, c2 = 0.0f;
        #pragma unroll
        for (int q = 0; q < 4; ++q) {
            int dd = lane + 32 * q;
            float e  = etile[m][dd];
            float cx = (cent_s[dd] - e) * (1.0f / (float)(NUTT - 1));
            dot += e * cx;
            e2  += e * e;
            c2  += cx * cx;
        }
        #pragma unroll
        for (int off = 16; off >= 1; off >>= 1) {
            dot += __shfl_xor(dot, off);
            e2  += __shfl_xor(e2,  off);
            c2  += __shfl_xor(c2,  off);
        }
        if (lane == 0) {
            float en = fmaxf(sqrtf(e2), EPSF);
            float cn = fmaxf(sqrtf(c2), EPSF);
            float sim = dot / (en * cn);
            int u = n * NUTT + m;
            own_logit[u] = W * sim + B;
            inv_en[u]    = 1.0f / en;
        }
    }
}

// ---------------------------------------------------------------------------
// Kernel 2: fused sim-matmul (v_wmma_f32_16x16x4_f32) + online log-softmax.
//   Each wave owns 16 utterance rows; loops 64 speaker tiles of 16.
//   D-tile layout: lane l -> utterance column (l%16); VGPR j -> speaker row
//   j + 8*(l/16). Lanes l and l^16 share a column -> one shfl merges halves.
// ---------------------------------------------------------------------------
__global__ __launch_bounds__(256)
void ge2e_main(const float* __restrict__ emb,
               const float* __restrict__ cent_hat,
               const float* __restrict__ inv_en,
               const float* __restrict__ own_logit,
               const float* __restrict__ wp, const float* __restrict__ bp,
               float* __restrict__ partials) {
    const int wave = (blockIdx.x * blockDim.x + threadIdx.x) >> 5;  // 0..2047
    const int l    = threadIdx.x & 31;
    const int col  = l & 15;             // utterance within tile
    const int kh   = l >> 4;             // K half: 0 or 1
    const int u    = wave * 16 + col;    // this lane's utterance row

    const float W    = wp[0];
    const float Boff = bp[0];
    const float inve = inv_en[u];
    const float ownl = own_logit[u];
    const int   nown = u >> 5;           // this utterance's speaker (uniform per wave)

    // B operand (K x 16 utterances), resident in 64 VGPRs for the whole kernel.
    // Lane holds emb[u][4t + 2*kh + {0,1}] for t = 0..31.
    v2f Breg[32];
    const float* erow = emb + (size_t)u * DIM + 2 * kh;
    #pragma unroll
    for (int t = 0; t < 32; ++t)
        Breg[t] = *(const v2f*)(erow + 4 * t);

    float M = -1e30f, S = 0.0f;          // online logsumexp state

    for (int st = 0; st < NSPK / 16; ++st) {
        v8f c = {0.f, 0.f, 0.f, 0.f, 0.f, 0.f, 0.f, 0.f};
        const float* arow = cent_hat + (size_t)(st * 16 + col) * DIM + 2 * kh;
        #pragma unroll
        for (int t = 0; t < 32; ++t) {
            v2f a = *(const v2f*)(arow + 4 * t);
            // D = A(16x4 f32) * B(4x16 f32) + C  -- v_wmma_f32_16x16x4_f32
            c = __builtin_amdgcn_wmma_f32_16x16x4_f32(
                    false, a, false, Breg[t], (short)0, c, false, false);
        }

        // logits for 8 speakers held by this lane; patch diagonal entry.
        float lg[8];
        float tm = -1e30f;
        #pragma unroll
        for (int j = 0; j < 8; ++j) {
            int spk = st * 16 + kh * 8 + j;
            float v = W * (c[j] * inve) + Boff;
            if (spk == nown) v = ownl;   // own-speaker uses exclusive centroid
            lg[j] = v;
            tm = fmaxf(tm, v);
        }
        tm = fmaxf(tm, __shfl_xor(tm, 16));      // merge speaker halves
        float ts = 0.0f;
        #pragma unroll
        for (int j = 0; j < 8; ++j) ts += __expf(lg[j] - tm);
        ts += __shfl_xor(ts, 16);

        float nM = fmaxf(M, tm);
        S = S * __expf(M - nM) + ts * __expf(tm - nM);
        M = nM;
    }

    float lse = M + __logf(S);
    float contrib = (kh == 0) ? (lse - ownl) : 0.0f;  // count each row once
    #pragma unroll
    for (int off = 16; off >= 1; off >>= 1) contrib += __shfl_xor(contrib, off);
    if (l == 0) partials[wave] = contrib;
}

// ---------------------------------------------------------------------------
// Kernel 3: deterministic reduction of 2048 wave partials -> mean loss.
// ---------------------------------------------------------------------------
__global__ __launch_bounds__(256)
void ge2e_finish(const float* __restrict__ partials, float* __restrict__ out) {
    __shared__ float red[256];
    float s = 0.0f;
    for (int i = threadIdx.x; i < NWAVE; i += 256) s += partials[i];
    red[threadIdx.x] = s;
    __syncthreads();
    for (int o = 128; o > 0; o >>= 1) {
        if (threadIdx.x < o) red[threadIdx.x] += red[threadIdx.x + o];
        __syncthreads();
    }
    if (threadIdx.x == 0) out[0] = red[0] / (float)NROW;
}

// ---------------------------------------------------------------------------
extern "C" void kernel_launch(void* const* d_in, const int* in_sizes, int n_in,
                              void* d_out, int out_size, void* d_ws, size_t ws_size,
                              hipStream_t stream) {
    const float* emb = (const float*)d_in[0];   // [32768, 128] f32
    const float* wp  = (const float*)d_in[1];   // scalar
    const float* bp  = (const float*)d_in[2];   // scalar
    // d_in[3], d_in[4]: n_speakers / n_utterances (compile-time constants here)

    // Workspace layout (f32):
    //   cent_hat : NSPK*DIM           = 131072 elems
    //   inv_en   : NROW               =  32768 elems
    //   own_logit: NROW               =  32768 elems
    //   partials : NWAVE              =   2048 elems
    float* ws        = (float*)d_ws;
    float* cent_hat  = ws;
    float* inv_en    = cent_hat + (size_t)NSPK * DIM;
    float* own_logit = inv_en + NROW;
    float* partials  = own_logit + NROW;

    ge2e_prep<<<NSPK, 128, 0, stream>>>(emb, wp, bp, cent_hat, inv_en, own_logit);
    ge2e_main<<<NWAVE / 8, 256, 0, stream>>>(emb, cent_hat, inv_en, own_logit,
                                             wp, bp, partials);
    ge2e_finish<<<1, 256, 0, stream>>>(partials, (float*)d_out);
}